// EdgeGraphAttention_30245159699048
// MI455X (gfx1250) — compile-verified
//
#include <hip/hip_runtime.h>
#include <hip/hip_bf16.h>

typedef float v2f __attribute__((ext_vector_type(2)));
typedef float v8f __attribute__((ext_vector_type(8)));

#define BATCH 16
#define NNODE 2000
#define NEDGE 8000
#define DFEAT 64

// ---------------------------------------------------------------------------
// Weight repack into WMMA-B native layout.
// For K-group g (4 K values), column col, lane-half t (t = 2*khalf + j):
//   P[((g*64 + col) << 2) + t] = W[(4g + t)*64 + col]
// so a lane's two B values (K = 4g+2*khalf, 4g+2*khalf+1) are one b64 load.
// ---------------------------------------------------------------------------
__global__ void pack_w(const float* __restrict__ W, float* __restrict__ P, int K)
{
    const int idx = blockIdx.x * blockDim.x + threadIdx.x;
    if (idx >= K * DFEAT) return;
    const int col = idx & 63;
    const int k   = idx >> 6;
    const int g   = k >> 2;
    const int t   = k & 3;
    P[((g * DFEAT + col) << 2) + t] = W[k * DFEAT + col];
}

// ---------------------------------------------------------------------------
// Kernel 1: Y[b,r,:] = X[b, rowOff + r, :] @ W (+ bias). One wave = 16x64 tile.
// A (16x4 f32): lanes 0-15 hold K=0,1 ; lanes 16-31 hold K=2,3.
// B (4x16 f32): same K split per lane-half; lane%16 = column (packed layout).
// D: VGPR v -> row v + 8*(lane>=16), col = lane%16.
// ---------------------------------------------------------------------------
__global__ void feat_gemm(const float* __restrict__ x, const float* __restrict__ Wp,
                          const float* __restrict__ bias, float* __restrict__ Y,
                          int rowsPerBatch, int rowOffsetInX, int useBias)
{
    const int wave  = blockIdx.x * (blockDim.x >> 5) + (threadIdx.x >> 5);
    const int lane  = threadIdx.x & 31;
    const int m     = lane & 15;
    const int khalf = lane >> 4;

    const long totalRows = (long)BATCH * rowsPerBatch;
    const long row0 = (long)wave * 16;
    if (row0 >= totalRows) return;

    const long gr = row0 + m;
    const int  b  = (int)(gr / rowsPerBatch);
    const int  r  = (int)(gr - (long)b * rowsPerBatch);
    const float* xrow = x + ((long)b * (NEDGE + NNODE) + rowOffsetInX + r) * DFEAT;
    const v2f* Bp = (const v2f*)Wp;   // v2f units: index = (g*64 + col)*2 + khalf

    v8f acc0 = {}, acc1 = {}, acc2 = {}, acc3 = {};

    #pragma unroll
    for (int kk = 0; kk < DFEAT; kk += 4) {
        const int g = kk >> 2;
        const v2f a = *(const v2f*)(xrow + kk + 2 * khalf);
        const v2f b0 = Bp[(g * DFEAT +  0 + m) * 2 + khalf];
        const v2f b1 = Bp[(g * DFEAT + 16 + m) * 2 + khalf];
        const v2f b2 = Bp[(g * DFEAT + 32 + m) * 2 + khalf];
        const v2f b3 = Bp[(g * DFEAT + 48 + m) * 2 + khalf];
        acc0 = __builtin_amdgcn_wmma_f32_16x16x4_f32(false, a, false, b0, (short)0, acc0, false, false);
        acc1 = __builtin_amdgcn_wmma_f32_16x16x4_f32(false, a, false, b1, (short)0, acc1, false, false);
        acc2 = __builtin_amdgcn_wmma_f32_16x16x4_f32(false, a, false, b2, (short)0, acc2, false, false);
        acc3 = __builtin_amdgcn_wmma_f32_16x16x4_f32(false, a, false, b3, (short)0, acc3, false, false);
    }

    #pragma unroll
    for (int v = 0; v < 8; ++v) {
        const int  md   = v + 8 * khalf;
        const long orow = row0 + md;
        float v0 = acc0[v], v1 = acc1[v], v2 = acc2[v], v3 = acc3[v];
        if (useBias) {
            v0 += bias[ 0 + m]; v1 += bias[16 + m];
            v2 += bias[32 + m]; v3 += bias[48 + m];
        }
        float* yrow = Y + orow * DFEAT;
        yrow[ 0 + m] = v0; yrow[16 + m] = v1;
        yrow[32 + m] = v2; yrow[48 + m] = v3;
    }
}

// ---------------------------------------------------------------------------
// Kernel 2: f_out[b,e,:] = leaky_relu([h[src]|f[eidx]|h[dst]] @ W_comb + b_comb)
// Gathered-A WMMA GEMM, K = 192 as three fully-unrolled 64-wide segments.
// Writes directly into d_out rows [0, Edir).
// ---------------------------------------------------------------------------
__global__ void comb_gemm(const float* __restrict__ h, const float* __restrict__ f,
                          const float* __restrict__ Wcp, const float* __restrict__ bc,
                          const int* __restrict__ src, const int* __restrict__ dst,
                          const int* __restrict__ eidx,
                          float* __restrict__ out, int Edir, int outRPB)
{
    const int wave  = blockIdx.x * (blockDim.x >> 5) + (threadIdx.x >> 5);
    const int lane  = threadIdx.x & 31;
    const int m     = lane & 15;
    const int khalf = lane >> 4;

    const long totalRows = (long)BATCH * Edir;
    const long row0 = (long)wave * 16;
    if (row0 >= totalRows) return;

    const long gr = row0 + m;
    const int  b  = (int)(gr / Edir);
    const int  e  = (int)(gr - (long)b * Edir);

    const int s  = src[e];
    const int d2 = dst[e];
    const int ei = eidx[e];
    const float* seg0 = h + ((long)b * NNODE + s ) * DFEAT;  // K   0..63
    const float* seg1 = f + ((long)b * NEDGE + ei) * DFEAT;  // K  64..127
    const float* seg2 = h + ((long)b * NNODE + d2) * DFEAT;  // K 128..191
    const v2f* Bp = (const v2f*)Wcp;  // v2f units: (gGlobal*64 + col)*2 + khalf

    v8f acc0 = {}, acc1 = {}, acc2 = {}, acc3 = {};

    #pragma unroll
    for (int seg = 0; seg < 3; ++seg) {
        const float* ap = (seg == 0) ? seg0 : ((seg == 1) ? seg1 : seg2);
        const int gbase = seg * (DFEAT / 4);
        #pragma unroll
        for (int kk = 0; kk < DFEAT; kk += 4) {
            const int g = gbase + (kk >> 2);
            const v2f a = *(const v2f*)(ap + kk + 2 * khalf);
            const v2f b0 = Bp[(g * DFEAT +  0 + m) * 2 + khalf];
            const v2f b1 = Bp[(g * DFEAT + 16 + m) * 2 + khalf];
            const v2f b2 = Bp[(g * DFEAT + 32 + m) * 2 + khalf];
            const v2f b3 = Bp[(g * DFEAT + 48 + m) * 2 + khalf];
            acc0 = __builtin_amdgcn_wmma_f32_16x16x4_f32(false, a, false, b0, (short)0, acc0, false, false);
            acc1 = __builtin_amdgcn_wmma_f32_16x16x4_f32(false, a, false, b1, (short)0, acc1, false, false);
            acc2 = __builtin_amdgcn_wmma_f32_16x16x4_f32(false, a, false, b2, (short)0, acc2, false, false);
            acc3 = __builtin_amdgcn_wmma_f32_16x16x4_f32(false, a, false, b3, (short)0, acc3, false, false);
        }
    }

    #pragma unroll
    for (int v = 0; v < 8; ++v) {
        const int  md   = v + 8 * khalf;
        const long orow = row0 + md;
        const int  ob   = (int)(orow / Edir);
        const int  oe   = (int)(orow - (long)ob * Edir);
        float v0 = acc0[v] + bc[ 0 + m];
        float v1 = acc1[v] + bc[16 + m];
        float v2 = acc2[v] + bc[32 + m];
        float v3 = acc3[v] + bc[48 + m];
        v0 = v0 > 0.f ? v0 : 0.01f * v0;
        v1 = v1 > 0.f ? v1 : 0.01f * v1;
        v2 = v2 > 0.f ? v2 : 0.01f * v2;
        v3 = v3 > 0.f ? v3 : 0.01f * v3;
        float* orowp = out + ((long)ob * outRPB + oe) * DFEAT;
        orowp[ 0 + m] = v0; orowp[16 + m] = v1;
        orowp[32 + m] = v2; orowp[48 + m] = v3;
    }
}

// ---------------------------------------------------------------------------
// Kernel 3: logits[b,e] = dot(f_out[b,e,:], w_attn). One wave per edge.
// ---------------------------------------------------------------------------
__global__ void attn_logits(const float* __restrict__ out, const float* __restrict__ w_attn,
                            float* __restrict__ logits, int Edir, int outRPB)
{
    const int wave = blockIdx.x * (blockDim.x >> 5) + (threadIdx.x >> 5);
    const int lane = threadIdx.x & 31;
    const long total = (long)BATCH * Edir;
    if (wave >= total) return;
    const int b = wave / Edir;
    const int e = wave - b * Edir;
    const float* row = out + ((long)b * outRPB + e) * DFEAT;
    float p = row[lane] * w_attn[lane] + row[lane + 32] * w_attn[lane + 32];
    #pragma unroll
    for (int off = 16; off > 0; off >>= 1) p += __shfl_xor(p, off, 32);
    if (lane == 0) logits[(long)b * Edir + e] = p;
}

// ---------------------------------------------------------------------------
// Kernel 4: in-place softmax over the Edir logits of each batch (1 block/batch)
// ---------------------------------------------------------------------------
__global__ void softmax_bn(float* __restrict__ logits, int Edir)
{
    __shared__ float red[256];
    __shared__ float s_max, s_sum;
    const int tid = threadIdx.x;
    float* L = logits + (long)blockIdx.x * Edir;

    float mx = -3.4e38f;
    for (int i = tid; i < Edir; i += 256) mx = fmaxf(mx, L[i]);
    red[tid] = mx; __syncthreads();
    for (int s = 128; s > 0; s >>= 1) {
        if (tid < s) red[tid] = fmaxf(red[tid], red[tid + s]);
        __syncthreads();
    }
    if (tid == 0) s_max = red[0];
    __syncthreads();
    const float mxv = s_max;

    float sum = 0.f;
    for (int i = tid; i < Edir; i += 256) sum += expf(L[i] - mxv);
    red[tid] = sum; __syncthreads();
    for (int s = 128; s > 0; s >>= 1) {
        if (tid < s) red[tid] += red[tid + s];
        __syncthreads();
    }
    if (tid == 0) s_sum = red[0];
    __syncthreads();
    const float inv = 1.f / s_sum;
    for (int i = tid; i < Edir; i += 256) L[i] = expf(L[i] - mxv) * inv;
}

// ---------------------------------------------------------------------------
// Kernel 5a/5b: CSR row bounds from the sorted src[] array.
// ---------------------------------------------------------------------------
__global__ void row_init(int* __restrict__ rs, int* __restrict__ re)
{
    const int n = blockIdx.x * blockDim.x + threadIdx.x;
    if (n < NNODE) { rs[n] = 0; re[n] = 0; }
}

__global__ void row_bounds(const int* __restrict__ src, int* __restrict__ rs,
                           int* __restrict__ re, int Edir)
{
    const int e = blockIdx.x * blockDim.x + threadIdx.x;
    if (e >= Edir) return;
    const int s = src[e];
    if (e == 0        || src[e - 1] != s) rs[s] = e;
    if (e == Edir - 1 || src[e + 1] != s) re[s] = e + 1;
}

// ---------------------------------------------------------------------------
// Kernel 6: h_out[b,n,:] = sum_{e in [rs[n],re[n])} alpha[b,e] * h[b,dst[e],:]
// Deterministic segmented reduction (src is sorted). One wave per (b,n).
// ---------------------------------------------------------------------------
__global__ void aggregate(const float* __restrict__ h, const float* __restrict__ alpha,
                          const int* __restrict__ dst, const int* __restrict__ rs,
                          const int* __restrict__ re, float* __restrict__ out,
                          int Edir, int outRPB)
{
    const int wave = blockIdx.x * (blockDim.x >> 5) + (threadIdx.x >> 5);
    const int lane = threadIdx.x & 31;
    const long total = (long)BATCH * NNODE;
    if (wave >= total) return;
    const int b = wave / NNODE;
    const int n = wave - b * NNODE;

    float a0 = 0.f, a1 = 0.f;
    const int s = rs[n], t = re[n];
    const float* hb = h + (long)b * NNODE * DFEAT;
    const float* al = alpha + (long)b * Edir;
    for (int e = s; e < t; ++e) {
        const float a = al[e];
        const float* hr = hb + (long)dst[e] * DFEAT;
        a0 += a * hr[lane];
        a1 += a * hr[lane + 32];
    }
    float* o = out + ((long)b * outRPB + Edir + n) * DFEAT;
    o[lane] = a0;
    o[lane + 32] = a1;
}

// ---------------------------------------------------------------------------
extern "C" void kernel_launch(void* const* d_in, const int* in_sizes, int n_in,
                              void* d_out, int out_size, void* d_ws, size_t ws_size,
                              hipStream_t stream)
{
    const float* x      = (const float*)d_in[0];
    const float* W_edge = (const float*)d_in[1];
    const float* W_node = (const float*)d_in[2];
    const float* b_node = (const float*)d_in[3];
    const float* W_comb = (const float*)d_in[4];
    const float* b_comb = (const float*)d_in[5];
    const float* w_attn = (const float*)d_in[6];
    const int*   src    = (const int*)d_in[7];
    const int*   dst    = (const int*)d_in[8];
    const int*   eidx   = (const int*)d_in[9];

    const int Edir   = in_sizes[7];          // directed edges (2 * NEDGE = 16000)
    const int outRPB = Edir + NNODE;         // output rows per batch

    // workspace carve (floats)
    float* f      = (float*)d_ws;                               // B*NEDGE*D
    float* h      = f + (size_t)BATCH * NEDGE * DFEAT;          // B*NNODE*D
    float* logits = h + (size_t)BATCH * NNODE * DFEAT;          // B*Edir (becomes alpha)
    float* pWe    = logits + (size_t)BATCH * Edir;              // 64*64 packed W_edge
    float* pWn    = pWe + (size_t)DFEAT * DFEAT;                // 64*64 packed W_node
    float* pWc    = pWn + (size_t)DFEAT * DFEAT;                // 192*64 packed W_comb
    int*   rs     = (int*)(pWc + (size_t)3 * DFEAT * DFEAT);    // NNODE
    int*   re     = rs + NNODE;                                 // NNODE

    const int threads = 256;                 // 8 waves/block
    const int wpb = threads / 32;

    // 0) repack weights into WMMA-B layout (tiny, L2-resident afterwards)
    pack_w<<<dim3((DFEAT * DFEAT + 255) / 256), dim3(256), 0, stream>>>(W_edge, pWe, DFEAT);
    pack_w<<<dim3((DFEAT * DFEAT + 255) / 256), dim3(256), 0, stream>>>(W_node, pWn, DFEAT);
    pack_w<<<dim3((3 * DFEAT * DFEAT + 255) / 256), dim3(256), 0, stream>>>(W_comb, pWc, 3 * DFEAT);

    // 1) f = edges @ W_edge
    {
        const long tiles = (long)BATCH * NEDGE / 16;
        feat_gemm<<<dim3((tiles + wpb - 1) / wpb), dim3(threads), 0, stream>>>(
            x, pWe, nullptr, f, NEDGE, 0, 0);
    }
    // 2) h = nodes @ W_node + b_node
    {
        const long tiles = (long)BATCH * NNODE / 16;
        feat_gemm<<<dim3((tiles + wpb - 1) / wpb), dim3(threads), 0, stream>>>(
            x, pWn, b_node, h, NNODE, NEDGE, 1);
    }
    // 3) f_out (into d_out) via gathered-A WMMA GEMM, K=192
    {
        const long tiles = ((long)BATCH * Edir + 15) / 16;
        comb_gemm<<<dim3((tiles + wpb - 1) / wpb), dim3(threads), 0, stream>>>(
            h, f, pWc, b_comb, src, dst, eidx, (float*)d_out, Edir, outRPB);
    }
    // 4) attention logits
    {
        const long waves = (long)BATCH * Edir;
        attn_logits<<<dim3((waves + wpb - 1) / wpb), dim3(threads), 0, stream>>>(
            (const float*)d_out, w_attn, logits, Edir, outRPB);
    }
    // 5) softmax per batch (in place -> alpha)
    softmax_bn<<<dim3(BATCH), dim3(256), 0, stream>>>(logits, Edir);

    // 6) CSR row bounds from sorted src
    row_init<<<dim3((NNODE + 255) / 256), dim3(256), 0, stream>>>(rs, re);
    row_bounds<<<dim3((Edir + 255) / 256), dim3(256), 0, stream>>>(src, rs, re, Edir);

    // 7) h_out aggregation (deterministic segmented reduction)
    {
        const long waves = (long)BATCH * NNODE;
        aggregate<<<dim3((waves + wpb - 1) / wpb), dim3(threads), 0, stream>>>(
            h, logits, dst, rs, re, (float*)d_out, Edir, outRPB);
    }
}